// IsingVectorModel_51462298140745
// MI455X (gfx1250) — compile-verified
//
#include <hip/hip_runtime.h>
#include <stdint.h>

// ---------------------------------------------------------------------------
// Ising vector model update for MI455X (gfx1250, wave32, WMMA).
//   N = 2048, D = 256, B = 4, PRIOR = 1.
//   bla == PRIOR^2 == 1 identically  =>  X = I - J shared across batches.
//   diag(inv(X)) via Newton-Schulz: Y <- 2Y - Y(XY), Y0 = X / (||X||_1^2)
//   (X symmetric => X^T = X, and X*Y0 is SPD => global convergence).
// Heavy math: V_WMMA_F32_16X16X4_F32.
// Data movement: GLOBAL_LOAD_ASYNC_TO_LDS_B128 + s_wait_asynccnt,
// double-buffered LDS so global->LDS latency overlaps the WMMA stream.
// ---------------------------------------------------------------------------

typedef float v2f __attribute__((ext_vector_type(2)));
typedef float v8f __attribute__((ext_vector_type(8)));

#define TILE_M 128
#define TILE_N 128
#define TILE_K 16
#define A_STR  20      // padded LDS stride for the A tile (16B-aligned b128 dests)
#define NS_ITERS 30

// ---- CDNA5 async global->LDS copy (ASYNCcnt-tracked, bypasses VGPRs) ------
__device__ __forceinline__ void async_load_b128(const float* gsrc, float* ldst) {
  uint32_t laddr = (uint32_t)(uintptr_t)ldst;   // low 32 bits = LDS byte address
  asm volatile("global_load_async_to_lds_b128 %0, %1, off"
               :: "v"(laddr), "v"(gsrc) : "memory");
}
__device__ __forceinline__ void wait_async_le4() {
  asm volatile("s_wait_asynccnt 0x4" ::: "memory");
}
__device__ __forceinline__ void wait_async_0() {
  asm volatile("s_wait_asynccnt 0x0" ::: "memory");
}

// ---------------------------------------------------------------- small ops
__global__ __launch_bounds__(256)
void build_JX_kernel(const float* __restrict__ Jraw, float* __restrict__ J,
                     float* __restrict__ X, int N) {
  int idx = blockIdx.x * blockDim.x + threadIdx.x;
  if (idx >= N * N) return;
  int i = idx / N;
  int j = idx - i * N;
  float v = (i == j) ? 0.0f : 0.5f * (Jraw[idx] + Jraw[(size_t)j * N + i]);
  J[idx] = v;
  X[idx] = ((i == j) ? 1.0f : 0.0f) - v;
}

__global__ __launch_bounds__(256)
void rowsum_abs_kernel(const float* __restrict__ X, float* __restrict__ rs, int N) {
  __shared__ float red[256];
  int row = blockIdx.x;
  float s = 0.0f;
  for (int j = threadIdx.x; j < N; j += 256)
    s += fabsf(X[(size_t)row * N + j]);
  red[threadIdx.x] = s;
  __syncthreads();
  for (int w = 128; w > 0; w >>= 1) {
    if (threadIdx.x < w) red[threadIdx.x] += red[threadIdx.x + w];
    __syncthreads();
  }
  if (threadIdx.x == 0) rs[row] = red[0];
}

__global__ __launch_bounds__(256)
void alpha_kernel(const float* __restrict__ rs, float* __restrict__ alpha, int N) {
  __shared__ float red[256];
  float m = 0.0f;
  for (int j = threadIdx.x; j < N; j += 256) m = fmaxf(m, rs[j]);
  red[threadIdx.x] = m;
  __syncthreads();
  for (int w = 128; w > 0; w >>= 1) {
    if (threadIdx.x < w) red[threadIdx.x] = fmaxf(red[threadIdx.x], red[threadIdx.x + w]);
    __syncthreads();
  }
  // X symmetric: ||X||_1 == ||X||_inf == max row abs-sum r.  alpha = 1/r^2.
  if (threadIdx.x == 0) alpha[0] = 1.0f / (red[0] * red[0]);
}

__global__ __launch_bounds__(256)
void scaleY0_kernel(const float* __restrict__ X, const float* __restrict__ alpha,
                    float* __restrict__ Y, int NN) {
  int idx = blockIdx.x * blockDim.x + threadIdx.x;
  if (idx < NN) Y[idx] = alpha[0] * X[idx];
}

__global__ __launch_bounds__(256)
void finalize_kernel(const float* __restrict__ cavity_var, const float* __restrict__ Y,
                     float* __restrict__ out_spin_var, float* __restrict__ out_cav_var,
                     int B, int N) {
  int idx = blockIdx.x * blockDim.x + threadIdx.x;
  if (idx >= B * N) return;
  int n = idx % N;
  float cv = cavity_var[idx];
  float pref = 1.0f / (1.0f - cv);                 // PRIOR^2 = 1
  float bla = cv + 1.0f / pref;                    // == 1 (matches reference fp path)
  out_spin_var[idx] = pref;
  out_cav_var[idx] = bla / Y[(size_t)n * N + n];   // bla / diag(inv(X))
}

// ---------------------------------------------------------------- WMMA GEMM
// modes: 0  C = A*B
//        1  C = 2*Cin - A*B                (Newton-Schulz step)
//        2  C = pref*(A*B - cv*sm + src)   (spin-mean update, batched over z)
__global__ __launch_bounds__(256)
void gemm_wmma_f32(const float* __restrict__ A, const float* __restrict__ Bm,
                   float* __restrict__ C, const float* __restrict__ Cin,
                   const float* __restrict__ cv, const float* __restrict__ sm,
                   const float* __restrict__ src,
                   int M, int Ncol, int K, int mode,
                   long long strideB, long long strideC) {
  __shared__ float As[2][TILE_M * A_STR];     // 2 x (128 x 16, stride 20)
  __shared__ float Bs[2][TILE_K * TILE_N];    // 2 x (16 x 128)

  const int z = blockIdx.z;
  const float* Bz  = Bm + (long long)z * strideB;
  float*       Cz  = C  + (long long)z * strideC;
  const float* cvz = cv  ? cv  + (long long)z * M       : nullptr;
  const float* smz = sm  ? sm  + (long long)z * strideC : nullptr;
  const float* srz = src ? src + (long long)z * strideC : nullptr;

  const int t    = threadIdx.x;
  const int lane = t & 31;
  const int wave = t >> 5;
  const int wm   = (wave >> 2) * 64;   // wave row band within 128
  const int wn   = (wave & 3) * 32;    // wave col band within 128
  const int l15  = lane & 15;
  const int h2   = (lane >> 4) * 2;    // K sub-pair select (A & B frags)
  const int h8   = (lane >> 4) * 8;    // C row-half select

  // staging coordinates (per thread: 2 b128 for A, 2 b128 for B per K-tile)
  const int ar = t >> 1;               // A tile: row 0..127
  const int ac = (t & 1) * 8;          // A tile: col 0 or 8
  const int brow0 = t >> 5;            // B tile rows via 512 float4 chunks
  const int bcol0 = (t & 31) * 4;
  const int brow1 = (t + 256) >> 5;
  const int bcol1 = ((t + 256) & 31) * 4;

  const float* gA = A + (size_t)(blockIdx.y * TILE_M + ar) * K + ac;
  const float* gB0 = Bz + (size_t)brow0 * Ncol + (size_t)blockIdx.x * TILE_N + bcol0;
  const float* gB1 = Bz + (size_t)brow1 * Ncol + (size_t)blockIdx.x * TILE_N + bcol1;

  float* lA0[2] = { &As[0][ar * A_STR + ac],          &As[1][ar * A_STR + ac] };
  float* lA1[2] = { &As[0][ar * A_STR + ac + 4],      &As[1][ar * A_STR + ac + 4] };
  float* lB0[2] = { &Bs[0][brow0 * TILE_N + bcol0],   &Bs[1][brow0 * TILE_N + bcol0] };
  float* lB1[2] = { &Bs[0][brow1 * TILE_N + bcol1],   &Bs[1][brow1 * TILE_N + bcol1] };

  v8f acc[4][2] = {};
  const int ktiles = K / TILE_K;

  // prologue: stage tile 0 into buffer 0
  {
    async_load_b128(gA, lA0[0]);
    async_load_b128(gA + 4, lA1[0]);
    async_load_b128(gB0, lB0[0]);
    async_load_b128(gB1, lB1[0]);
  }

  for (int kt = 0; kt < ktiles; ++kt) {
    const int p = kt & 1;
    if (kt + 1 < ktiles) {
      // stage tile kt+1 into the other buffer while we compute this one
      const size_t koffA = (size_t)(kt + 1) * TILE_K;
      const size_t koffB = (size_t)(kt + 1) * TILE_K * Ncol;
      async_load_b128(gA + koffA, lA0[p ^ 1]);
      async_load_b128(gA + koffA + 4, lA1[p ^ 1]);
      async_load_b128(gB0 + koffB, lB0[p ^ 1]);
      async_load_b128(gB1 + koffB, lB1[p ^ 1]);
      wait_async_le4();   // in-order completion: current buffer's 4 ops done
    } else {
      wait_async_0();
    }
    __syncthreads();      // current buffer visible to all waves

    const float* __restrict__ Ap = As[p];
    const float* __restrict__ Bp = Bs[p];
#pragma unroll
    for (int ks = 0; ks < TILE_K; ks += 4) {
      const int koff = ks + h2;
      v2f af[4];
      v2f bf[2];
#pragma unroll
      for (int mi = 0; mi < 4; ++mi) {
        const float* pA = &Ap[(wm + mi * 16 + l15) * A_STR + koff];
        af[mi].x = pA[0];
        af[mi].y = pA[1];
      }
#pragma unroll
      for (int ni = 0; ni < 2; ++ni) {
        bf[ni].x = Bp[koff * TILE_N + wn + ni * 16 + l15];
        bf[ni].y = Bp[(koff + 1) * TILE_N + wn + ni * 16 + l15];
      }
#pragma unroll
      for (int mi = 0; mi < 4; ++mi)
#pragma unroll
        for (int ni = 0; ni < 2; ++ni)
          acc[mi][ni] = __builtin_amdgcn_wmma_f32_16x16x4_f32(
              false, af[mi], false, bf[ni], (short)0, acc[mi][ni], false, false);
    }
    __syncthreads();      // everyone done reading buffer p before it refills
  }

  // ---- epilogue
#pragma unroll
  for (int mi = 0; mi < 4; ++mi) {
#pragma unroll
    for (int ni = 0; ni < 2; ++ni) {
      const int col = blockIdx.x * TILE_N + wn + ni * 16 + l15;
#pragma unroll
      for (int r = 0; r < 8; ++r) {
        const int row = blockIdx.y * TILE_M + wm + mi * 16 + r + h8;
        const size_t off = (size_t)row * Ncol + col;
        float v = acc[mi][ni][r];
        if (mode == 0) {
          Cz[off] = v;
        } else if (mode == 1) {
          Cz[off] = 2.0f * Cin[off] - v;
        } else {
          float c = cvz[row];
          float pref = 1.0f / (1.0f - c);
          Cz[off] = pref * (v - c * smz[off] + srz[off]);
        }
      }
    }
  }
}

// ---------------------------------------------------------------- launcher
extern "C" void kernel_launch(void* const* d_in, const int* in_sizes, int n_in,
                              void* d_out, int out_size, void* d_ws, size_t ws_size,
                              hipStream_t stream) {
  (void)in_sizes; (void)n_in; (void)out_size; (void)ws_size;
  const int N = 2048, D = 256, B = 4;
  const long long NN = (long long)N * N;

  const float* spin_mean  = (const float*)d_in[0];
  // d_in[1] (spin_var) is unused by the reference
  const float* cavity_var = (const float*)d_in[2];
  const float* source     = (const float*)d_in[3];
  const float* J_raw      = (const float*)d_in[4];

  float* out = (float*)d_out;
  float* out_spin_mean = out;                             // B*N*D
  float* out_spin_var  = out + (long long)B * N * D;      // B*N
  float* out_cav_var   = out_spin_var + (long long)B * N; // B*N

  float* ws = (float*)d_ws;
  float* J  = ws;
  float* X  = ws + NN;
  float* Y  = ws + 2 * NN;
  float* Z  = ws + 3 * NN;
  float* T  = ws + 4 * NN;
  float* rs = ws + 5 * NN;
  float* al = ws + 5 * NN + N;

  // 1) J = sym_traceless(J_raw);  X = I - J
  build_JX_kernel<<<(int)(NN / 256), 256, 0, stream>>>(J_raw, J, X, N);

  // 2) Newton-Schulz seed:  Y0 = X / (max row abs-sum)^2
  rowsum_abs_kernel<<<N, 256, 0, stream>>>(X, rs, N);
  alpha_kernel<<<1, 256, 0, stream>>>(rs, al, N);
  scaleY0_kernel<<<(int)(NN / 256), 256, 0, stream>>>(X, al, Y, (int)NN);

  // 3) Fused spin-mean update (batched over z):
  //    out = pref * (J @ spin_mean - cv*spin_mean + source)
  {
    dim3 grid(D / TILE_N, N / TILE_M, B);
    gemm_wmma_f32<<<grid, 256, 0, stream>>>(
        J, spin_mean, out_spin_mean, nullptr, cavity_var, spin_mean, source,
        N, D, N, /*mode=*/2, (long long)N * D, (long long)N * D);
  }

  // 4) Newton-Schulz:  Z = X*Ycur ;  Ynext = 2*Ycur - Ycur*Z ; swap
  float* Ycur = Y;
  float* Yalt = T;
  dim3 gNN(N / TILE_N, N / TILE_M, 1);
  for (int it = 0; it < NS_ITERS; ++it) {
    gemm_wmma_f32<<<gNN, 256, 0, stream>>>(
        X, Ycur, Z, nullptr, nullptr, nullptr, nullptr,
        N, N, N, /*mode=*/0, 0LL, 0LL);
    gemm_wmma_f32<<<gNN, 256, 0, stream>>>(
        Ycur, Z, Yalt, Ycur, nullptr, nullptr, nullptr,
        N, N, N, /*mode=*/1, 0LL, 0LL);
    float* tmp = Ycur; Ycur = Yalt; Yalt = tmp;
  }

  // 5) next_spin_var = pref ;  next_cavity_var = bla / diag(inv(X))
  finalize_kernel<<<(B * N + 255) / 256, 256, 0, stream>>>(
      cavity_var, Ycur, out_spin_var, out_cav_var, B, N);
}